// FixedDGCNNBackbone_45397804319293
// MI455X (gfx1250) — compile-verified
//
#include <hip/hip_runtime.h>
#include <hip/hip_bf16.h>

typedef float v2f __attribute__((ext_vector_type(2)));
typedef float v8f __attribute__((ext_vector_type(8)));

#define KNN 20
#define BN_EPS 1e-5f
#define NEG_INF -3.402823466e38f
#define POS_INF  3.402823466e38f

// Compiler-only memory barrier: forces LDS values to be re-loaded (same-wave
// DS ops complete in order, so no hardware fence is needed for intra-wave
// cross-lane visibility through LDS).
#define LDS_COMPILER_FENCE() __asm__ __volatile__("" ::: "memory")

// D = A(16x4, f32) * B(4x16, f32) + C(16x16, f32), wave32.
// Lane layout (per CDNA5 ISA 7.12.2):
//   A: lane l -> M = l&15, VGPR0/1 = K = (l<16 ? 0,1 : 2,3)
//   B: lane l -> N = l&15, VGPR0/1 = K = (l<16 ? 0,1 : 2,3)
//   C/D: VGPR r -> M = r + (l<16 ? 0 : 8), N = l&15
__device__ __forceinline__ v8f wmma_f32x4(v2f a, v2f b, v8f c) {
  return __builtin_amdgcn_wmma_f32_16x16x4_f32(false, a, false, b, (short)0, c,
                                               false, false);
}

__device__ __forceinline__ float leaky02(float x) { return fmaxf(x, 0.2f * x); }

// -------------------------------------------------------------------------
// xx[b,n] = sum_c X[b,c,n]^2
__global__ __launch_bounds__(256) void norms_kernel(
    const float* __restrict__ X, int strideB, int C, int N, int B,
    float* __restrict__ xx) {
  int id = blockIdx.x * 256 + threadIdx.x;
  if (id >= B * N) return;
  int b = id / N, n = id - b * N;
  const float* Xb = X + (size_t)b * strideB;
  float s = 0.0f;
  for (int c = 0; c < C; ++c) {
    float v = Xb[c * N + n];
    s += v * v;
  }
  xx[id] = s;
}

// -------------------------------------------------------------------------
// Exact top-K (K=20) of pd[n,m] = 2*dot(x_n,x_m) - ||x_n||^2 - ||x_m||^2
// (negative squared distance, self included — matches reference).
// One workgroup = 16 rows (points). 8 waves, each computing 16x16 f32 WMMA
// Gram tiles over disjoint column tiles, maintaining per-wave top-20 lists
// per row in LDS, then an 8-way merge.
__global__ __launch_bounds__(256) void knn_kernel(
    const float* __restrict__ X, int strideB, int C, int N,
    const float* __restrict__ xx, int* __restrict__ idxOut) {
  __shared__ float As[128 * 16];            // row-block features, zero-padded
  __shared__ float lval[8 * 16 * KNN];      // per-wave top-20 values
  __shared__ int   lcol[8 * 16 * KNN];      // per-wave top-20 indices

  const int tid  = threadIdx.x;
  const int w    = tid >> 5;
  const int lane = tid & 31;
  const bool hi  = lane >= 16;
  const int colN = lane & 15;
  const int kb   = hi ? 2 : 0;
  const int b    = blockIdx.y;
  const int n0   = blockIdx.x * 16;
  const int CP    = (C + 3) & ~3;   // padded K extent
  const int Cmain = C & ~3;         // branch-free K extent

  const float* Xb  = X + (size_t)b * strideB;
  const float* xxb = xx + (size_t)b * N;

  for (int i = tid; i < 8 * 16 * KNN; i += 256) { lval[i] = NEG_INF; lcol[i] = 0; }
  for (int i = tid; i < CP * 16; i += 256) {
    int c = i >> 4, m = i & 15;
    As[i] = (c < C) ? Xb[c * N + (n0 + m)] : 0.0f;
  }
  __syncthreads();

  float xxr[8];
#pragma unroll
  for (int r = 0; r < 8; ++r) xxr[r] = xxb[n0 + r + (hi ? 8 : 0)];

  const int ntiles = N >> 4;
  for (int mt = w; mt < ntiles; mt += 8) {
    const int m0 = mt << 4;
    const float* Xcol = Xb + (m0 + colN);   // column for this lane's B frags
    v8f acc = {};
    // main K loop: branch-free (C is a multiple of 4 for layers 2-4)
    for (int c0 = 0; c0 < Cmain; c0 += 4) {
      const int c1 = c0 + kb;
      v2f a, bf;
      a.x  = As[c1 * 16 + colN];
      a.y  = As[(c1 + 1) * 16 + colN];
      bf.x = Xcol[c1 * N];
      bf.y = Xcol[(c1 + 1) * N];
      acc = wmma_f32x4(a, bf, acc);
    }
    // tail chunk (only layer 1, C=3): zero-pad B loads past C
    if (Cmain < CP) {
      const int c1 = Cmain + kb, c2 = Cmain + kb + 1;
      v2f a, bf;
      a.x  = As[c1 * 16 + colN];
      a.y  = As[(c1 + 1) * 16 + colN];
      bf.x = (c1 < C) ? Xcol[c1 * N] : 0.0f;
      bf.y = (c2 < C) ? Xcol[c2 * N] : 0.0f;
      acc = wmma_f32x4(a, bf, acc);
    }
    const float xxc = xxb[m0 + colN];
#pragma unroll
    for (int r = 0; r < 8; ++r) {
      const int row   = r + (hi ? 8 : 0);
      const int lbase = (w * 16 + row) * KNN;
      const float cand = 2.0f * acc[r] - xxr[r] - xxc;
      const int   col  = m0 + colN;
      bool done = false;
      // ballot-serialized exact insertion, threshold-filtered
      while (true) {
        LDS_COMPILER_FENCE();
        float thr = lval[lbase + KNN - 1];
        unsigned msk = (unsigned)__ballot((!done) && (cand > thr));
        if (msk == 0u) break;
        int leader = __ffs(msk) - 1;
        if (lane == leader) {
          int j = KNN - 1;
          while (j > 0 && cand > lval[lbase + j - 1]) {
            lval[lbase + j] = lval[lbase + j - 1];
            lcol[lbase + j] = lcol[lbase + j - 1];
            --j;
          }
          lval[lbase + j] = cand;
          lcol[lbase + j] = col;
          done = true;
        }
        LDS_COMPILER_FENCE();
      }
    }
  }
  __syncthreads();

  if (tid < 16) {  // merge 8 sorted per-wave lists for row `tid`
    float mv[KNN]; int mc[KNN];
#pragma unroll
    for (int k = 0; k < KNN; ++k) { mv[k] = NEG_INF; mc[k] = 0; }
    for (int wv = 0; wv < 8; ++wv) {
      int base = (wv * 16 + tid) * KNN;
      for (int e = 0; e < KNN; ++e) {
        float cv = lval[base + e];
        if (cv <= mv[KNN - 1]) break;  // lists are sorted descending
        int cc = lcol[base + e];
        int j = KNN - 1;
        while (j > 0 && cv > mv[j - 1]) { mv[j] = mv[j - 1]; mc[j] = mc[j - 1]; --j; }
        mv[j] = cv; mc[j] = cc;
      }
    }
    size_t obase = ((size_t)b * N + (n0 + tid)) * KNN;
    for (int k = 0; k < KNN; ++k) idxOut[obase + k] = mc[k];
  }
}

// -------------------------------------------------------------------------
// V[b,o,n] = sum_c Wd[o,c]*X[b,c,n],  U[b,o,n] = sum_c (Wc-Wd)[o,c]*X[b,c,n]
// W is [O][2C]; Wd = W[:, :C], Wc = W[:, C:]. Both GEMMs share B fragments.
__global__ __launch_bounds__(128) void gemm_vu_kernel(
    const float* __restrict__ W, int C, const float* __restrict__ X,
    int xStrideB, int N, float* __restrict__ V, float* __restrict__ U, int O) {
  const int tid  = threadIdx.x;
  const int w    = tid >> 5;
  const int lane = tid & 31;
  const bool hi  = lane >= 16;
  const int colN = lane & 15;
  const int kb   = hi ? 2 : 0;
  const int b    = blockIdx.z;
  const int o0   = blockIdx.y * 16;
  const int n0   = (blockIdx.x * 4 + w) * 16;
  const int CP    = (C + 3) & ~3;
  const int Cmain = C & ~3;
  const int twoC  = 2 * C;

  const float* wrow = W + (o0 + colN) * twoC;
  const float* Xcol = X + (size_t)b * xStrideB + (n0 + colN);
  v8f accV = {}, accU = {};
  for (int c0 = 0; c0 < Cmain; c0 += 4) {
    const int c1 = c0 + kb;
    const float wd0 = wrow[c1];
    const float wd1 = wrow[c1 + 1];
    const float wc0 = wrow[C + c1];
    const float wc1 = wrow[C + c1 + 1];
    v2f aV; aV.x = wd0;       aV.y = wd1;
    v2f aU; aU.x = wc0 - wd0; aU.y = wc1 - wd1;
    v2f bf; bf.x = Xcol[c1 * N]; bf.y = Xcol[(c1 + 1) * N];
    accV = wmma_f32x4(aV, bf, accV);
    accU = wmma_f32x4(aU, bf, accU);
  }
  if (Cmain < CP) {  // tail chunk (layer 1, C=3)
    const int c1 = Cmain + kb, c2 = Cmain + kb + 1;
    const float wd0 = (c1 < C) ? wrow[c1] : 0.0f;
    const float wd1 = (c2 < C) ? wrow[c2] : 0.0f;
    const float wc0 = (c1 < C) ? wrow[C + c1] : 0.0f;
    const float wc1 = (c2 < C) ? wrow[C + c2] : 0.0f;
    v2f aV; aV.x = wd0;       aV.y = wd1;
    v2f aU; aU.x = wc0 - wd0; aU.y = wc1 - wd1;
    v2f bf;
    bf.x = (c1 < C) ? Xcol[c1 * N] : 0.0f;
    bf.y = (c2 < C) ? Xcol[c2 * N] : 0.0f;
    accV = wmma_f32x4(aV, bf, accV);
    accU = wmma_f32x4(aU, bf, accU);
  }
  float* Vb = V + ((size_t)b * O + o0 + (hi ? 8 : 0)) * N + (n0 + colN);
  float* Ub = U + ((size_t)b * O + o0 + (hi ? 8 : 0)) * N + (n0 + colN);
#pragma unroll
  for (int r = 0; r < 8; ++r) {
    Vb[r * N] = accV[r];
    Ub[r * N] = accU[r];
  }
}

// -------------------------------------------------------------------------
// out[b,o,n] = LeakyReLU( s[o]*(ext_k(V[b,o,idx[n,k]]) + U[b,o,n]) + t[o] )
// ext = max if s>=0 else min (max over k commutes through monotone BN+leaky).
__global__ __launch_bounds__(256) void gather_max_kernel(
    const float* __restrict__ V, const float* __restrict__ U,
    const int* __restrict__ idx, const float* __restrict__ bnp,
    float* __restrict__ out, int outStrideB, int O, int N) {
  __shared__ int sidx[32 * KNN];
  const int tid = threadIdx.x;
  const int b   = blockIdx.y;
  const int n0  = blockIdx.x * 32;
  for (int i = tid; i < 32 * KNN; i += 256)
    sidx[i] = idx[((size_t)b * N + n0) * KNN + i];
  __syncthreads();
  const int p  = tid & 31;
  const int og = tid >> 5;
  for (int o = og; o < O; o += 8) {
    const float g = bnp[o], beta = bnp[O + o];
    const float mean = bnp[2 * O + o], var = bnp[3 * O + o];
    const float s = g * rsqrtf(var + BN_EPS);
    const float t = beta - mean * s;
    const float* vrow = V + ((size_t)b * O + o) * N;
    float m;
    if (s >= 0.0f) {
      m = NEG_INF;
      for (int k = 0; k < KNN; ++k) m = fmaxf(m, vrow[sidx[p * KNN + k]]);
    } else {
      m = POS_INF;
      for (int k = 0; k < KNN; ++k) m = fminf(m, vrow[sidx[p * KNN + k]]);
    }
    const float z = m + U[((size_t)b * O + o) * N + (n0 + p)];
    out[(size_t)b * outStrideB + o * N + (n0 + p)] = leaky02(s * z + t);
  }
}

// -------------------------------------------------------------------------
// Final conv5 (512x512) + BN + LeakyReLU, fused epilogue.
__global__ __launch_bounds__(128) void conv5_kernel(
    const float* __restrict__ W, const float* __restrict__ cat,
    const float* __restrict__ bnp, float* __restrict__ out, int N) {
  const int C = 512, O = 512;
  const int tid  = threadIdx.x;
  const int w    = tid >> 5;
  const int lane = tid & 31;
  const bool hi  = lane >= 16;
  const int colN = lane & 15;
  const int kb   = hi ? 2 : 0;
  const int b    = blockIdx.z;
  const int o0   = blockIdx.y * 16;
  const int n0   = (blockIdx.x * 4 + w) * 16;

  const float* wrow = W + (o0 + colN) * C;
  const float* Xcol = cat + (size_t)b * C * N + (n0 + colN);
  v8f acc = {};
  for (int c0 = 0; c0 < C; c0 += 4) {
    const int c1 = c0 + kb;
    v2f a;  a.x  = wrow[c1];
            a.y  = wrow[c1 + 1];
    v2f bf; bf.x = Xcol[c1 * N];
            bf.y = Xcol[(c1 + 1) * N];
    acc = wmma_f32x4(a, bf, acc);
  }
  float* ob = out + ((size_t)b * O + o0 + (hi ? 8 : 0)) * N + (n0 + colN);
  const int obn = o0 + (hi ? 8 : 0);
#pragma unroll
  for (int r = 0; r < 8; ++r) {
    const int o = obn + r;
    const float g = bnp[o], beta = bnp[O + o];
    const float mean = bnp[2 * O + o], var = bnp[3 * O + o];
    const float s = g * rsqrtf(var + BN_EPS);
    const float t = beta - mean * s;
    ob[r * N] = leaky02(s * acc[r] + t);
  }
}

// -------------------------------------------------------------------------
extern "C" void kernel_launch(void* const* d_in, const int* in_sizes, int n_in,
                              void* d_out, int out_size, void* d_ws,
                              size_t ws_size, hipStream_t stream) {
  (void)in_sizes; (void)n_in; (void)out_size; (void)ws_size;
  const int B = 2, N = 8192;
  const float* x   = (const float*)d_in[0];
  const float* W1  = (const float*)d_in[1];
  const float* W2  = (const float*)d_in[2];
  const float* W3  = (const float*)d_in[3];
  const float* W4  = (const float*)d_in[4];
  const float* W5  = (const float*)d_in[5];
  const float* bn1 = (const float*)d_in[6];
  const float* bn2 = (const float*)d_in[7];
  const float* bn3 = (const float*)d_in[8];
  const float* bn4 = (const float*)d_in[9];
  const float* bn5 = (const float*)d_in[10];

  // workspace carve (floats): cat[512*B*N] | xx[B*N] | idx[B*N*K] | V | U
  float* cat = (float*)d_ws;                        // 32 MB, conv5 input layout
  float* xx  = cat + (size_t)512 * B * N;           // point norms
  int*   idx = (int*)(xx + (size_t)B * N);          // top-K indices
  float* V   = (float*)(idx + (size_t)B * N * KNN); // 16 MB (O<=256)
  float* U   = V + (size_t)256 * B * N;             // 16 MB

  struct Layer {
    const float* in; int strideB; int C;
    const float* W; const float* bn; int O; int ch0;
  };
  const Layer L[4] = {
      { x,             3 * N,   3,   W1, bn1, 64,  0   },
      { cat,           512 * N, 64,  W2, bn2, 64,  64  },
      { cat + 64 * N,  512 * N, 64,  W3, bn3, 128, 128 },
      { cat + 128 * N, 512 * N, 128, W4, bn4, 256, 256 },
  };

  for (int l = 0; l < 4; ++l) {
    norms_kernel<<<dim3((B * N + 255) / 256), dim3(256), 0, stream>>>(
        L[l].in, L[l].strideB, L[l].C, N, B, xx);
    knn_kernel<<<dim3(N / 16, B), dim3(256), 0, stream>>>(
        L[l].in, L[l].strideB, L[l].C, N, xx, idx);
    gemm_vu_kernel<<<dim3(N / 64, L[l].O / 16, B), dim3(128), 0, stream>>>(
        L[l].W, L[l].C, L[l].in, L[l].strideB, N, V, U, L[l].O);
    gather_max_kernel<<<dim3(N / 32, B), dim3(256), 0, stream>>>(
        V, U, idx, L[l].bn, cat + (size_t)L[l].ch0 * N, 512 * N, L[l].O, N);
  }
  conv5_kernel<<<dim3(N / 64, 512 / 16, B), dim3(128), 0, stream>>>(
      W5, cat, bn5, (float*)d_out, N);
}